// SavgolPreprocessor_48567490183685
// MI455X (gfx1250) — compile-verified
//
#include <hip/hip_runtime.h>
#include <math.h>

// CDNA5 / gfx1250 Savitzky-Golay preprocessor.
// One workgroup (8 wave32) per batch row (T=2048).
// - async global->LDS staging of x,y rows (ASYNCcnt path)
// - 7-tap convs done as banded matmul via V_WMMA_F32_16X16X4_F32
// - unwrap+D1 conv reformulated as a purely local stencil (D1 sums to zero)
// - weights synthesized with branchless integer ALU (no tables, no branches)

#define T_LEN 2048
#define PI_F 3.14159265358979323846f
#define TWO_PI_F 6.28318530717958647692f

typedef __attribute__((ext_vector_type(2))) float v2f;
typedef __attribute__((ext_vector_type(4))) float v4f;
typedef __attribute__((ext_vector_type(8))) float v8f;
typedef int b128_t __attribute__((vector_size(16)));          // matches builtin param
typedef __attribute__((address_space(1))) b128_t* gptr_b128;  // global (AS1)
typedef __attribute__((address_space(3))) b128_t* lptr_b128;  // LDS (AS3)

__device__ __forceinline__ int iclamp(int v, int lo, int hi) {
    return v < lo ? lo : (v > hi ? hi : v);
}

// D1[d] = (3-d)/28 for d in [0,6], else 0.  Branchless.
__device__ __forceinline__ float w_d1(int d) {
    const int n = ((unsigned)d <= 6u) ? (3 - d) : 0;
    return (float)n * (1.0f / 28.0f);
}

// D2[d] = [5,1,-2,-4,-2,1,5]/42.  Numerator = (q^2 + 3|q| - 8)/2, q = d-3.  Branchless.
__device__ __forceinline__ float w_d2(int d) {
    const int q  = d - 3;
    const int qa = q < 0 ? -q : q;
    int n = (q * q + 3 * qa - 8) >> 1;     // |q|=0..3 -> -4,-2,1,5
    n = ((unsigned)d <= 6u) ? n : 0;
    return (float)n * (1.0f / 42.0f);
}

extern "C" __global__ __launch_bounds__(256)
void savgol_wmma_kernel(const float* __restrict__ xy, float* __restrict__ out)
{
    __shared__ __align__(16) float ldx[T_LEN];
    __shared__ __align__(16) float ldy[T_LEN];
    __shared__ __align__(16) float ldth[T_LEN];

    const int b   = blockIdx.x;
    const int tid = threadIdx.x;

    const float* __restrict__ xrow = xy + (size_t)b * (2 * T_LEN);
    const float* __restrict__ yrow = xrow + T_LEN;

    // ---- Stage x,y rows into LDS -------------------------------------------
#if defined(__gfx1250__) && __has_builtin(__builtin_amdgcn_global_load_async_to_lds_b128)
    #pragma unroll
    for (int j = 0; j < 2; ++j) {
        const int v4 = tid + 256 * j;   // float4 index, 512 per row
        __builtin_amdgcn_global_load_async_to_lds_b128(
            (gptr_b128)(xrow + 4 * v4),
            (lptr_b128)(&ldx[4 * v4]), 0, 0);
        __builtin_amdgcn_global_load_async_to_lds_b128(
            (gptr_b128)(yrow + 4 * v4),
            (lptr_b128)(&ldy[4 * v4]), 0, 0);
    }
    #if __has_builtin(__builtin_amdgcn_s_wait_asynccnt)
        __builtin_amdgcn_s_wait_asynccnt(0);
    #else
        asm volatile("s_wait_asynccnt 0" ::: "memory");
    #endif
#else
    #pragma unroll
    for (int j = 0; j < 2; ++j) {
        const int v4 = tid + 256 * j;
        ((v4f*)ldx)[v4] = ((const v4f*)xrow)[v4];
        ((v4f*)ldy)[v4] = ((const v4f*)yrow)[v4];
    }
#endif
    __syncthreads();

    // ---- Band-matrix conv via WMMA f32 16x16x4 -----------------------------
    // Wave tile: 256 consecutive t. C[M][N], M = offset in 16-subtile,
    // N = subtile index. C = W(16x24) x Xwin(24x16), split into 6 K=4 chunks.
    const int lane     = tid & 31;
    const int wave     = tid >> 5;
    const int waveBase = wave * 256;
    const int N        = lane & 15;   // B/C/D column, A row (M = lane%16)
    const int hi       = lane >> 4;   // lane half
    const int colBase  = waveBase + 16 * N;

    v8f accDX  = {0.f,0.f,0.f,0.f,0.f,0.f,0.f,0.f};
    v8f accDY  = {0.f,0.f,0.f,0.f,0.f,0.f,0.f,0.f};
    v8f accD2X = {0.f,0.f,0.f,0.f,0.f,0.f,0.f,0.f};
    v8f accD2Y = {0.f,0.f,0.f,0.f,0.f,0.f,0.f,0.f};

    #pragma unroll
    for (int c = 0; c < 6; ++c) {
        // Per-lane K indices for this chunk (A layout: VGPRv, lane half -> K = 4c + v + 2*hi)
        const int kA = 4 * c + 2 * hi;
        const int d0 = kA - N;          // W[i][k] = D[k-i], i == lane%16
        const int d1 = d0 + 1;

        v2f a1, a2, bx, by;
        a1.x = w_d1(d0);  a1.y = w_d1(d1);
        a2.x = w_d2(d0);  a2.y = w_d2(d1);

        // B layout assumed symmetric to A: VGPRv, lane half -> K = 4c + v + 2*hi
        const int i0 = iclamp(colBase + kA - 3, 0, T_LEN - 1);
        const int i1 = iclamp(colBase + kA - 2, 0, T_LEN - 1);
        bx.x = ldx[i0]; bx.y = ldx[i1];
        by.x = ldy[i0]; by.y = ldy[i1];

        accDX  = __builtin_amdgcn_wmma_f32_16x16x4_f32(false, a1, false, bx, (short)0, accDX,  false, false);
        accDY  = __builtin_amdgcn_wmma_f32_16x16x4_f32(false, a1, false, by, (short)0, accDY,  false, false);
        accD2X = __builtin_amdgcn_wmma_f32_16x16x4_f32(false, a2, false, bx, (short)0, accD2X, false, false);
        accD2Y = __builtin_amdgcn_wmma_f32_16x16x4_f32(false, a2, false, by, (short)0, accD2Y, false, false);
    }

    // ---- theta = atan2(dy,dx) into LDS (C/D layout: t = base+16N + r + 8*hi)
    #pragma unroll
    for (int r = 0; r < 8; ++r) {
        const int t = colBase + r + 8 * hi;
        ldth[t] = atan2f(accDY[r], accDX[r]);
    }
    __syncthreads();

    // ---- Fused epilogue: speed, local-unwrap curvature, channel-last store --
    constexpr float D1t[7] = { 0.10714286f,  0.07142857f,  0.03571429f, 0.0f,
                              -0.03571429f, -0.07142857f, -0.10714286f };
    float* __restrict__ orow = out + (size_t)b * (T_LEN * 8);

    #pragma unroll
    for (int r = 0; r < 8; ++r) {
        const int t = colBase + r + 8 * hi;
        const float dx  = accDX[r],  dy  = accDY[r];
        const float d2x = accD2X[r], d2y = accD2Y[r];
        const float speed = sqrtf(dx * dx + dy * dy + 1e-8f);

        // curvature = clip( conv(unwrap(theta), D1), -2, 2 )
        // local form: sum_j D1[j]*(th_j + S_j), S_j = running +-2pi corrections
        float th_prev = ldth[iclamp(t - 3, 0, T_LEN - 1)];
        float S = 0.0f;
        float curv = D1t[0] * th_prev;
        #pragma unroll
        for (int j = 1; j < 7; ++j) {
            const float th = ldth[iclamp(t + j - 3, 0, T_LEN - 1)];
            const float d = th - th_prev;
            S += (d > PI_F) ? -TWO_PI_F : ((d < -PI_F) ? TWO_PI_F : 0.0f);
            curv += D1t[j] * (th + S);
            th_prev = th;
        }
        curv = fminf(fmaxf(curv, -2.0f), 2.0f);

        const v4f o0 = { ldx[t], ldy[t], dx, dy };
        const v4f o1 = { d2x, d2y, speed, curv };
        v4f* op = (v4f*)(orow + (size_t)t * 8);
        __builtin_nontemporal_store(o0, op);
        __builtin_nontemporal_store(o1, op + 1);
    }
}

extern "C" void kernel_launch(void* const* d_in, const int* in_sizes, int n_in,
                              void* d_out, int out_size, void* d_ws, size_t ws_size,
                              hipStream_t stream) {
    (void)n_in; (void)out_size; (void)d_ws; (void)ws_size;
    const float* xy = (const float*)d_in[0];
    float* out = (float*)d_out;
    const int B = in_sizes[0] / (2 * T_LEN);   // 4096 for the reference shapes
    savgol_wmma_kernel<<<B, 256, 0, stream>>>(xy, out);
}